// VariantCoeLinear1d_82291573391982
// MI455X (gfx1250) — compile-verified
//
#include <hip/hip_runtime.h>
#include <cstdint>
#include <cstddef>

typedef float v2f __attribute__((ext_vector_type(2)));

namespace {
constexpr int kB = 128;
constexpr int kN = 2048;
constexpr int kTPB = 256;              // 8 wave32s per workgroup
constexpr int kChunk = kN / kTPB;      // 8 contiguous nodes per thread
constexpr int kPairs = kChunk / 2;     // 4 float2 pairs per thread
}

__device__ __forceinline__ v2f splat2(float x) {
  v2f r; r[0] = x; r[1] = x; return r;
}

// f(u) = 1.5u + 18.75u^2 - 50.5u^3 + 37.5u^4 - 6.25u^6   (packed Horner)
__device__ __forceinline__ v2f f_real_v2(v2f u) {
  v2f p = __builtin_elementwise_fma(splat2(-6.25f) * u, u, splat2(37.5f));
  p = __builtin_elementwise_fma(p, u, splat2(-50.5f));
  p = __builtin_elementwise_fma(p, u, splat2(18.75f));
  p = __builtin_elementwise_fma(p, u, splat2(1.5f));
  return p * u;
}

// f'(u) = 1.5 + 37.5u - 151.5u^2 + 150u^3 - 37.5u^5   (no abs here; folded later)
__device__ __forceinline__ v2f dfdu_v2(v2f u) {
  v2f q = splat2(-37.5f) * u;
  q = __builtin_elementwise_fma(q, u, splat2(150.0f));
  q = __builtin_elementwise_fma(q, u, splat2(-151.5f));
  q = __builtin_elementwise_fma(q, u, splat2(37.5f));
  q = __builtin_elementwise_fma(q, u, splat2(1.5f));
  return q;
}

// alpha = max(|qL|,|qR|): abs folds into v_max input modifiers
__device__ __forceinline__ float alpha_max(float qL, float qR) {
  return fmaxf(fabsf(qL), fabsf(qR));
}

__global__ __launch_bounds__(kTPB) void lax_friedrichs_rows(
    const float* __restrict__ init, const int* __restrict__ stepnum_p,
    float* __restrict__ out) {
  __shared__ __align__(16) float u0_s[kN];   // double-buffered state
  __shared__ __align__(16) float u1_s[kN];

  const int steps = *stepnum_p;          // uniform scalar load
  const int b = blockIdx.x;              // one batch row per workgroup
  const int tid = threadIdx.x;
  const int base = tid * kChunk;
  const bool first = (tid == 0);
  const bool last = (tid == kTPB - 1);
  const float hl = 0.5f * (float)(5.0e-4 / (10.0 / 2048.0));  // lam/2 (G = 2F)

  // Load this thread's 8 nodes of the initial condition (2x b128) into pairs.
  v2f uu[kPairs];
  {
    const float4* src = (const float4*)(init + (size_t)b * kN + base);
    float4 v0 = src[0];
    float4 v1 = src[1];
    uu[0][0] = v0.x; uu[0][1] = v0.y; uu[1][0] = v0.z; uu[1][1] = v0.w;
    uu[2][0] = v1.x; uu[2][1] = v1.y; uu[3][0] = v1.z; uu[3][1] = v1.w;
  }

  float* w = u0_s;                                         // written this step
  float* r = u1_s;                                         // written next step
  unsigned ldsw = (unsigned)(uintptr_t)(u0_s + base);      // LDS byte offsets
  unsigned ldso = (unsigned)(uintptr_t)(u1_s + base);
  unsigned long long g =
      (unsigned long long)(uintptr_t)(out + (size_t)b * kN + base);
  const unsigned long long gstep = (unsigned long long)kB * kN * sizeof(float);

  for (int t = 0; t < steps - 1; ++t) {
    // Async stores reading buffer `w` were issued 2 steps ago; allow only the
    // previous step's 2 stores to remain in flight before rewriting `w`.
    asm volatile("s_wait_asynccnt 0x2" ::: "memory");

    // ---- publish u^t to LDS buffer w ----
    ((float4*)(w + base))[0] =
        make_float4(uu[0][0], uu[0][1], uu[1][0], uu[1][1]);
    ((float4*)(w + base))[1] =
        make_float4(uu[2][0], uu[2][1], uu[3][0], uu[3][1]);
    __syncthreads();   // single barrier per step (double buffering)

    // ---- stream u^t -> out[t] via CDNA5 async LDS->global DMA (ASYNCcnt) ----
    asm volatile("global_store_async_from_lds_b128 %0, %1, off"
                 :: "v"(g), "v"(ldsw) : "memory");
    asm volatile("global_store_async_from_lds_b128 %0, %1, off offset:16"
                 :: "v"(g), "v"(ldsw) : "memory");

    // ---- halo u from LDS (one packed chain evaluates both halos) ----
    v2f h;
    h[0] = first ? 0.f : w[base - 1];         // u[base-1]
    h[1] = last ? 0.f : w[base + kChunk];     // u[base+8]

    float ue[kChunk], fe[kChunk], qe[kChunk];
#pragma unroll
    for (int p = 0; p < kPairs; ++p) {
      v2f fp = f_real_v2(uu[p]);              // v_pk_fma_f32 chains
      v2f qp = dfdu_v2(uu[p]);
      ue[2 * p] = uu[p][0]; ue[2 * p + 1] = uu[p][1];
      fe[2 * p] = fp[0];    fe[2 * p + 1] = fp[1];
      qe[2 * p] = qp[0];    qe[2 * p + 1] = qp[1];
    }
    const v2f fh = f_real_v2(h);
    const v2f qh = dfdu_v2(h);

    // G[i] = doubled flux at interface (base+i-1, base+i), i = 0..8
    float G[kChunk + 1];
    G[0] = first ? 0.f
                 : (fh[0] + fe[0]) - alpha_max(qh[0], qe[0]) * (ue[0] - h[0]);
#pragma unroll
    for (int i = 1; i < kChunk; ++i)
      G[i] = (fe[i - 1] + fe[i]) -
             alpha_max(qe[i - 1], qe[i]) * (ue[i] - ue[i - 1]);
    G[kChunk] = last ? 0.f
                     : (fe[kChunk - 1] + fh[1]) -
                           alpha_max(qe[kChunk - 1], qh[1]) * (h[1] - ue[kChunk - 1]);

    float nu[kChunk];
#pragma unroll
    for (int m = 0; m < kChunk; ++m)
      nu[m] = fmaf(-hl, G[m + 1] - G[m], ue[m]);
    if (first) nu[0] = nu[1];                    // u[:,0]   = u[:,1]
    if (last) nu[kChunk - 1] = nu[kChunk - 2];   // u[:,N-1] = u[:,N-2]
#pragma unroll
    for (int p = 0; p < kPairs; ++p) {
      uu[p][0] = nu[2 * p];
      uu[p][1] = nu[2 * p + 1];
    }

    // swap buffers; advance output row pointer by one step (1 MB)
    float* tf = w; w = r; r = tf;
    unsigned tl = ldsw; ldsw = ldso; ldso = tl;
    g += gstep;
  }

  // ---- final state u^{steps-1} -> out[steps-1] ----
  asm volatile("s_wait_asynccnt 0x2" ::: "memory");
  ((float4*)(w + base))[0] = make_float4(uu[0][0], uu[0][1], uu[1][0], uu[1][1]);
  ((float4*)(w + base))[1] = make_float4(uu[2][0], uu[2][1], uu[3][0], uu[3][1]);
  __syncthreads();
  asm volatile("global_store_async_from_lds_b128 %0, %1, off"
               :: "v"(g), "v"(ldsw) : "memory");
  asm volatile("global_store_async_from_lds_b128 %0, %1, off offset:16"
               :: "v"(g), "v"(ldsw) : "memory");
  asm volatile("s_wait_asynccnt 0x0" ::: "memory");
}

extern "C" void kernel_launch(void* const* d_in, const int* in_sizes, int n_in,
                              void* d_out, int out_size, void* d_ws, size_t ws_size,
                              hipStream_t stream) {
  const float* init = (const float*)d_in[0];   // [128, 2048] f32
  const int* stepnum = (const int*)d_in[1];    // scalar int (500)
  float* out = (float*)d_out;                  // [500, 128, 2048] f32
  (void)in_sizes; (void)n_in; (void)out_size; (void)d_ws; (void)ws_size;

  lax_friedrichs_rows<<<dim3(kB), dim3(kTPB), 0, stream>>>(init, stepnum, out);
}